// TopographicIntentMap_70841190580502
// MI455X (gfx1250) — compile-verified
//
#include <hip/hip_runtime.h>

typedef float v2f __attribute__((ext_vector_type(2)));
typedef float v8f __attribute__((ext_vector_type(8)));

#define NUMN 144      // neurons
#define NUME 115      // excitatory
#define NB   8        // neighbors
#define NTIL 9        // N tiles of 16 (144)
#define KTIL 36       // K tiles of 4  (144)
#define ROWS_PER_WG 128
#define WAVES 8

// ---------------------------------------------------------------------------
// Kernel A: const_drive[b,o] = 0.5*action_current - 0.5 * sum_e S[b,e]*W_inh[o,e]
// where S[b,e] = sum_n spikes[b,n,e].  One block per batch row.
// ---------------------------------------------------------------------------
__global__ __launch_bounds__(160) void drive_kernel(const int* __restrict__ actions,
                                                    const float* __restrict__ spikes,
                                                    const float* __restrict__ Winh,
                                                    float* __restrict__ drive)
{
    __shared__ float S[NUME];
    const int b   = blockIdx.x;
    const int tid = threadIdx.x;

    for (int e = tid; e < NUME; e += 160) {
        const float* sp = spikes + (size_t)b * NB * NUME + e;
        float s = 0.f;
#pragma unroll
        for (int n = 0; n < NB; ++n) s += sp[n * NUME];
        S[e] = s;
    }
    __syncthreads();

    if (tid < NUMN) {
        const int a    = actions[b];
        // action -> grid cell: 0:(1,0)->5  1:(0,-1)->1  2:(-1,0)->3  3:(0,1)->7  4:(0,0)->4
        const int cell = (a == 0) ? 5 : (a == 1) ? 1 : (a == 2) ? 3 : (a == 3) ? 7 : 4;
        const float* wr = Winh + tid * NUME;
        float acc = 0.f;
        for (int e = 0; e < NUME; ++e) acc = fmaf(S[e], wr[e], acc);
        const float inp = ((tid >> 4) == cell) ? 5.0f : 0.0f;
        drive[(size_t)b * NUMN + tid] = 0.5f * inp - 0.5f * acc;
    }
}

// ---------------------------------------------------------------------------
// Kernel B: 30-tick LIF recurrence.  8 waves/block, 16 batch rows per wave.
// rec = acc @ W_rec^T via V_WMMA_F32_16X16X4_F32, W_rec + per-wave acc staging
// in LDS.  State (v, acc) lives in registers in WMMA C-layout.
// ---------------------------------------------------------------------------
__global__ __launch_bounds__(256) void sim_kernel(const float* __restrict__ drive,
                                                  const float* __restrict__ Wrec,
                                                  const int* __restrict__ ticks,
                                                  float* __restrict__ out)
{
    extern __shared__ float smem[];
    float* Wr = smem;                                   // [144][144] row-major

    const int tid      = threadIdx.x;
    const int wave     = tid >> 5;
    const int lane     = tid & 31;
    const int laneCol  = lane & 15;                     // N (or M for A-frag) index
    const int laneHalf = lane >> 4;                     // 0 or 1
    float* accS = smem + NUMN * NUMN + wave * 16 * NUMN; // wave-private [16][144]

    const int row0 = blockIdx.x * ROWS_PER_WG + wave * 16;

    // Cooperative load of W_rec into LDS (read as B-matrix: B[k][n] = Wrec[n][k]).
    for (int i = tid; i < NUMN * NUMN; i += 256) Wr[i] = Wrec[i];
    __syncthreads();

    const v8f vzero = {0.f, 0.f, 0.f, 0.f, 0.f, 0.f, 0.f, 0.f};
    v8f vstate[NTIL];
    v8f accv[NTIL];
#pragma unroll
    for (int n = 0; n < NTIL; ++n) { vstate[n] = vzero; accv[n] = vzero; }

    const int T = *ticks;
    const float* drvBase = drive + (size_t)(row0 + 8 * laneHalf) * NUMN + laneCol;
    const float* aBase   = accS + laneCol * NUMN + 2 * laneHalf;
    const float* bBase   = Wr   + laneCol * NUMN + 2 * laneHalf;

    for (int t = 0; t < T; ++t) {
        // ---- stage acc (C-layout regs) into row-major [m][k] LDS tile -----
#pragma unroll
        for (int n = 0; n < NTIL; ++n) {
#pragma unroll
            for (int r = 0; r < 8; ++r)
                accS[(r + 8 * laneHalf) * NUMN + n * 16 + laneCol] = accv[n][r];
        }

        // ---- rec = acc @ W_rec^T : 36 K-steps of 16x16x4 f32 WMMA ---------
        v8f rec[NTIL];
#pragma unroll
        for (int n = 0; n < NTIL; ++n) rec[n] = vzero;

        for (int k = 0; k < KTIL; ++k) {
            // A fragment: lane m = laneCol, K pair starting at 4k + 2*laneHalf
            v2f a = *(const v2f*)(aBase + k * 4);
#pragma unroll
            for (int n = 0; n < NTIL; ++n) {
                // B fragment: B[kk][n] = Wrec[n*16+laneCol][4k + 2*laneHalf + {0,1}]
                v2f bb = *(const v2f*)(bBase + n * 16 * NUMN + k * 4);
                rec[n] = __builtin_amdgcn_wmma_f32_16x16x4_f32(
                    false, a, false, bb, (short)0, rec[n], false, false);
            }
        }

        // ---- LIF: x = drive + 0.3*ei*rec ; v=(v+x)/2 ; spike/reset --------
#pragma unroll
        for (int n = 0; n < NTIL; ++n) {
            const float eim = ((n * 16 + laneCol) < NUME) ? 0.3f : -0.3f;
#pragma unroll
            for (int r = 0; r < 8; ++r) {
                const float d  = drvBase[(size_t)r * NUMN + n * 16];
                const float x  = fmaf(eim, rec[n][r], d);
                const float vv = 0.5f * (vstate[n][r] + x);
                const bool fire = (vv >= 1.0f);
                vstate[n][r] = fire ? 0.0f : vv;
                accv[n][r]  += fire ? 1.0f : 0.0f;
            }
        }
    }

    // ---- write acc[:, :115] ----------------------------------------------
#pragma unroll
    for (int n = 0; n < NTIL; ++n) {
        const int col = n * 16 + laneCol;
        if (col < NUME) {
#pragma unroll
            for (int r = 0; r < 8; ++r)
                out[(size_t)(row0 + r + 8 * laneHalf) * NUME + col] = accv[n][r];
        }
    }
}

// ---------------------------------------------------------------------------
extern "C" void kernel_launch(void* const* d_in, const int* in_sizes, int n_in,
                              void* d_out, int out_size, void* d_ws, size_t ws_size,
                              hipStream_t stream)
{
    const int*   actions = (const int*)  d_in[0];
    const float* spikes  = (const float*)d_in[1];
    const float* Wrec    = (const float*)d_in[2];
    const float* Winh    = (const float*)d_in[3];
    const int*   ticks   = (const int*)  d_in[4];
    float*       outp    = (float*)d_out;
    float*       drive   = (float*)d_ws;            // B * 144 floats of scratch

    const int B = in_sizes[0];                      // 16384

    drive_kernel<<<B, 160, 0, stream>>>(actions, spikes, Winh, drive);

    const size_t lds = (size_t)(NUMN * NUMN + WAVES * 16 * NUMN) * sizeof(float); // ~153 KB
    sim_kernel<<<B / ROWS_PER_WG, 256, lds, stream>>>(drive, Wrec, ticks, outp);

    (void)n_in; (void)out_size; (void)ws_size;
}